// RippleNet_35442070126903
// MI455X (gfx1250) — compile-verified
//
#include <hip/hip_runtime.h>
#include <hip/hip_bf16.h>
#include <math.h>

// Problem constants (match reference)
#define HOPS 2
#define BATCH 2048
#define MEM 64
#define DIM 16
#define NREL 32

typedef __attribute__((ext_vector_type(2))) float v2f;
typedef __attribute__((ext_vector_type(8))) float v8f;

// ---------------------------------------------------------------------------
// Kernel 1: q[b, r, :] = item_e[items[b]]^T @ rel_emb[r]   for all r in [0,32)
// One wave computes a 16(items) x 16(j) tile for each relation r, accumulating
// a 16x16x16 f32 GEMM tile as 4 chained V_WMMA_F32_16X16X4_F32 ops.
// rel_emb (32 KB) is staged in LDS once per block; the A operand (items tile)
// is loaded once per wave and reused across all 32 relations.
// Block = 128 threads (4 waves) -> 64 item rows per block -> grid = B/64 = 32.
// Also emits the gathered item embeddings to ws for kernel 2's final dot.
// ---------------------------------------------------------------------------
__global__ __launch_bounds__(128)
void ripple_itemR_wmma(const int* __restrict__ items,
                       const float* __restrict__ ent_emb,
                       const float* __restrict__ rel_emb,
                       float* __restrict__ q,       // [B][NREL][DIM]
                       float* __restrict__ itemv)   // [B][DIM]
{
    __shared__ float relLDS[NREL * DIM * DIM];   // 32 KB
    __shared__ float itemLDS[64 * DIM];          // 4 KB

    const int tid = threadIdx.x;
    const int rowbase = blockIdx.x * 64;

    // Stage all 32 relation matrices into LDS (8192 floats / 128 threads).
    #pragma unroll
    for (int i = tid; i < NREL * DIM * DIM; i += 128)
        relLDS[i] = rel_emb[i];

    // Gather this block's 64 item embeddings into LDS + workspace.
    for (int i = tid; i < 64 * DIM; i += 128) {
        const int row = i >> 4;
        const int j   = i & 15;
        const float v = ent_emb[(size_t)items[rowbase + row] * DIM + j];
        itemLDS[i] = v;
        itemv[(size_t)(rowbase + row) * DIM + j] = v;
    }
    __syncthreads();

    const int wave  = tid >> 5;       // 0..3 : which 16-item tile
    const int lane  = tid & 31;
    const int mrow  = lane & 15;      // M row (lane 0-15) / also N col for B
    const int khalf = lane >> 4;      // 0: K{0,1}, 1: K{2,3} for A; row sel for B
    const int trow  = wave * 16;

    // A operand: 16x4 f32 per K-chunk, 2 VGPRs per lane (ISA 7.12.2).
    // lane<16: {K=4kc+0, K=4kc+1}; lane>=16: {K=4kc+2, K=4kc+3}, same M row.
    v2f a[4];
    #pragma unroll
    for (int kc = 0; kc < 4; ++kc) {
        const int kb = 4 * kc + 2 * khalf;
        a[kc].x = itemLDS[(trow + mrow) * DIM + kb + 0];
        a[kc].y = itemLDS[(trow + mrow) * DIM + kb + 1];
    }

    for (int r = 0; r < NREL; ++r) {
        const float* R = &relLDS[r * DIM * DIM];   // R[i][j], row-major
        v8f c = {};   // zero-init accumulator
        #pragma unroll
        for (int kc = 0; kc < 4; ++kc) {
            // B operand: 4x16 f32, rows striped across half-waves:
            // v0: K = 4kc + khalf ; v1: K = 4kc + 2 + khalf ; N = lane&15.
            v2f bm;
            bm.x = R[(4 * kc + khalf)     * DIM + mrow];
            bm.y = R[(4 * kc + 2 + khalf) * DIM + mrow];
            // 8 args: (neg_a, A, neg_b, B, c_mod, C, reuse_a, reuse_b)
            c = __builtin_amdgcn_wmma_f32_16x16x4_f32(
                    false, a[kc], false, bm, (short)0, c, false, false);
        }
        // D layout: VGPR v, lane L -> M = v + 8*(L>=16), N = L&15.
        #pragma unroll
        for (int v = 0; v < 8; ++v) {
            const int row = trow + v + 8 * khalf;
            q[((size_t)(rowbase + row) * NREL + r) * DIM + mrow] = c[v];
        }
    }
}

// ---------------------------------------------------------------------------
// Kernel 2: one block (128 threads) per batch element b.
// Thread t handles slot (h = t>>6, m = t&63):
//   logit = dot(q[b, rel[h,b,m]], ent[head[h,b,m]])      (16-dim dot)
// softmax over m per hop, then user_rep = sum_{h,m} pi * tail_e, then
// out[b] = sigmoid(dot(user_rep, item_e[b])).
// ---------------------------------------------------------------------------
__global__ __launch_bounds__(128)
void ripple_attend(const int* __restrict__ heads,
                   const int* __restrict__ rels,
                   const int* __restrict__ tails,
                   const float* __restrict__ ent_emb,
                   const float* __restrict__ q,      // [B][NREL][DIM]
                   const float* __restrict__ itemv,  // [B][DIM]
                   float* __restrict__ out)          // [B]
{
    __shared__ float logits[HOPS * MEM];      // 128
    __shared__ float pi_s[HOPS * MEM];        // 128
    __shared__ float tail_s[HOPS * MEM][DIM]; // 8 KB
    __shared__ float part[8][DIM];            // partial tail sums

    const int b   = blockIdx.x;
    const int tid = threadIdx.x;          // 0..127
    const int h   = tid >> 6;
    const int m   = tid & 63;
    const size_t slot = ((size_t)h * BATCH + b) * MEM + m;

    const int r  = rels[slot];
    const int hd = heads[slot];
    const int tl = tails[slot];

    const float* qv = q + ((size_t)b * NREL + r) * DIM;
    const float* hv = ent_emb + (size_t)hd * DIM;
    float lg = 0.f;
    #pragma unroll
    for (int j = 0; j < DIM; ++j) lg += qv[j] * hv[j];
    logits[tid] = lg;

    const float* tv = ent_emb + (size_t)tl * DIM;
    #pragma unroll
    for (int j = 0; j < DIM; ++j) tail_s[tid][j] = tv[j];
    __syncthreads();

    // Softmax over the 64 slots of this hop (small M: direct LDS scan).
    const int base = h * MEM;
    float mx = -INFINITY;
    #pragma unroll 8
    for (int i = 0; i < MEM; ++i) mx = fmaxf(mx, logits[base + i]);
    float se = 0.f;
    #pragma unroll 8
    for (int i = 0; i < MEM; ++i) se += __expf(logits[base + i] - mx);
    pi_s[tid] = __expf(lg - mx) / se;
    __syncthreads();

    // user_rep[d] = sum over all 128 slots of pi * tail. Thread t reduces
    // d = t&15 over its 16-slot chunk, then threads 0..15 fold 8 partials.
    {
        const int d  = tid & 15;
        const int ch = tid >> 4;
        float p = 0.f;
        #pragma unroll
        for (int s = 0; s < 16; ++s) {
            const int sl = ch * 16 + s;
            p += pi_s[sl] * tail_s[sl][d];
        }
        part[ch][d] = p;
    }
    __syncthreads();

    if (tid < DIM) {
        float acc = 0.f;
        #pragma unroll
        for (int c2 = 0; c2 < 8; ++c2) acc += part[c2][tid];
        part[0][tid] = acc * itemv[(size_t)b * DIM + tid];
    }
    __syncthreads();

    if (tid == 0) {
        float s = 0.f;
        #pragma unroll
        for (int j = 0; j < DIM; ++j) s += part[0][j];
        out[b] = 1.f / (1.f + __expf(-s));
    }
}

// ---------------------------------------------------------------------------
// Launch
// ---------------------------------------------------------------------------
extern "C" void kernel_launch(void* const* d_in, const int* in_sizes, int n_in,
                              void* d_out, int out_size, void* d_ws, size_t ws_size,
                              hipStream_t stream) {
    const int*   items    = (const int*)  d_in[0];   // [B]
    const int*   heads    = (const int*)  d_in[1];   // [H,B,M]
    const int*   rels     = (const int*)  d_in[2];   // [H,B,M]
    const int*   tails    = (const int*)  d_in[3];   // [H,B,M]
    const float* ent_emb  = (const float*)d_in[4];   // [N_ENT, D]
    const float* rel_emb  = (const float*)d_in[5];   // [N_REL, D, D]
    float*       out      = (float*)d_out;           // [B]

    // Workspace layout: q [B][32][16] f32 (4 MB), itemv [B][16] f32 (128 KB).
    float* q     = (float*)d_ws;
    float* itemv = q + (size_t)BATCH * NREL * DIM;

    ripple_itemR_wmma<<<BATCH / 64, 128, 0, stream>>>(items, ent_emb, rel_emb, q, itemv);
    ripple_attend<<<BATCH, 128, 0, stream>>>(heads, rels, tails, ent_emb, q, itemv, out);
}